// SDPool_86079734547153
// MI455X (gfx1250) — compile-verified
//
#include <hip/hip_runtime.h>
#include <hip/hip_bf16.h>

// SDPool forward for MI455X (gfx1250, wave32, WMMA).
// All matmuls use V_WMMA_F32_16X16X4_F32 (fp32 matrix core) since the
// spectral-distance argmin is precision-sensitive. Big GEMMs stage panels in
// LDS via the CDNA5 async load-to-LDS path (ASYNCcnt) when available.
// Eigenvalues via parallel Householder tridiagonalization + tql1 per matrix.

typedef float v2f __attribute__((ext_vector_type(2)));
typedef float v8f __attribute__((ext_vector_type(8)));
typedef int   v4i __attribute__((ext_vector_type(4)));

#define DEVINL __device__ __forceinline__

constexpr int cB = 32;      // graphs
constexpr int cN = 512;     // nodes per graph
constexpr int cK = 64;      // landmark size
constexpr int cD = 128;     // hidden
constexpr int cL = 4;       // landmark graphs
constexpr int cT = cB * cN; // 16384

constexpr long POOLSEL_OFF = (long)cB * cK * cD;               // 262144
constexpr long SPMIN_OFF   = POOLSEL_OFF + (long)cB * cK * cK; // 393216

// ---------------- CDNA5 async global->LDS copy (guarded; safe fallback) ----
#if defined(__has_builtin)
#if __has_builtin(__builtin_amdgcn_global_load_async_to_lds_b128)
#define HAVE_ASYNC_LDS 1
#endif
#endif
#ifndef HAVE_ASYNC_LDS
#define HAVE_ASYNC_LDS 0
#endif

#if HAVE_ASYNC_LDS
#if __has_builtin(__builtin_amdgcn_s_wait_asynccnt)
#define ASYNC_WAIT() __builtin_amdgcn_s_wait_asynccnt(0)
#else
#define ASYNC_WAIT() asm volatile("s_wait_asynccnt 0x0" ::: "memory")
#endif
#else
#define ASYNC_WAIT()
#endif

typedef __attribute__((address_space(1))) v4i* gv4i_p;  // global int4*
typedef __attribute__((address_space(3))) v4i* lv4i_p;  // LDS int4*

DEVINL void cp16(void* lds, const void* g) {
#if HAVE_ASYNC_LDS
  __builtin_amdgcn_global_load_async_to_lds_b128((gv4i_p)g, (lv4i_p)lds, 0, 0);
#else
  *(float4*)lds = *(const float4*)g;
#endif
}

// ---------------------------------------------------------------- reductions
DEVINL float blk_sum256(float v, float* red) {
  int tid = threadIdx.x;
  red[tid] = v;
  __syncthreads();
  for (int s = 128; s > 0; s >>= 1) {
    if (tid < s) red[tid] += red[tid + s];
    __syncthreads();
  }
  float r = red[0];
  __syncthreads();
  return r;
}

// ---------------------------------------------------------------- WMMA core
// One wave computes a 16x16 fp32 tile of C = A*B, stepping K by 4 with
// V_WMMA_F32_16X16X4_F32.  32-bit A 16x4 layout: lanes 0-15 hold M=lane,
// K={0,1} in v{0,1}; lanes 16-31 hold M=lane-16, K={2,3}.  B mirrors with N
// striped across lanes.  OPA/OPB==1 => operand stored transposed.
template<int OPA, int OPB>
DEVINL v8f wmma_mm(const float* __restrict__ A, int lda,
                   const float* __restrict__ Bm, int ldb,
                   int m0, int n0, int Kd)
{
  const int lane = threadIdx.x & 31;
  const int half = lane >> 4;
  const int lr   = lane & 15;
  v8f acc = {0.f,0.f,0.f,0.f,0.f,0.f,0.f,0.f};
  __builtin_prefetch(&A[OPA ? (long)(m0 + lr) : (long)(m0 + lr) * lda], 0, 1);
  __builtin_prefetch(&Bm[OPB ? (long)(n0 + lr) * ldb : (long)(n0 + lr)], 0, 1);
  for (int k = 0; k < Kd; k += 4) {
    int k0 = k + (half << 1);
    v2f av, bv;
    if (OPA) {            // strided: two b32
      av.x = A[(long)k0 * lda + (m0 + lr)];
      av.y = A[(long)(k0 + 1) * lda + (m0 + lr)];
    } else {              // contiguous: one b64
      av = *(const v2f*)&A[(long)(m0 + lr) * lda + k0];
    }
    if (OPB) {
      bv = *(const v2f*)&Bm[(long)(n0 + lr) * ldb + k0];
    } else {
      bv.x = Bm[(long)k0 * ldb + (n0 + lr)];
      bv.y = Bm[(long)(k0 + 1) * ldb + (n0 + lr)];
    }
    acc = __builtin_amdgcn_wmma_f32_16x16x4_f32(false, av, false, bv,
                                                (short)0, acc, false, false);
  }
  return acc;
}

// EPI: 0 = none, 1 = relu, 2 = zero the matrix diagonal
template<int EPI>
DEVINL void wmma_store(float* __restrict__ C, int ldc, int m0, int n0, v8f acc)
{
  const int lane = threadIdx.x & 31;
  const int half = lane >> 4;
  const int lr   = lane & 15;
#pragma unroll
  for (int r = 0; r < 8; ++r) {
    int row = m0 + r + (half << 3);
    int col = n0 + lr;
    float v = acc[r];
    if (EPI == 1) v = v > 0.f ? v : 0.f;
    if (EPI == 2 && row == col) v = 0.f;
    C[(long)row * ldc + col] = v;
  }
}

// --------------------- direct batched GEMM (small/irregular operands) ------
template<int OPA, int OPB, int EPI>
__global__ void gemm_f32(const float* __restrict__ A, long sA, int aMod, int lda,
                         const float* __restrict__ Bm, long sB, int bMod, int ldb,
                         float* __restrict__ C, long sC, int ldc,
                         int M, int Nn, int Kd, int nbatch)
{
  int wid = blockIdx.x * (blockDim.x >> 5) + (threadIdx.x >> 5);
  int tilesM = M >> 4, tilesN = Nn >> 4;
  long per = (long)tilesM * tilesN;
  if (wid >= per * nbatch) return;
  int bat = (int)(wid / per);
  int t   = (int)(wid - (long)bat * per);
  int tm  = t / tilesN;
  int tn  = t - tm * tilesN;
  const float* Ab = A  + (long)(bat % aMod) * sA;
  const float* Bb = Bm + (long)(bat % bMod) * sB;
  float*       Cb = C  + (long)bat * sC;
  v8f acc = wmma_mm<OPA, OPB>(Ab, lda, Bb, ldb, tm << 4, tn << 4, Kd);
  wmma_store<EPI>(Cb, ldc, tm << 4, tn << 4, acc);
}

// --------------- LDS-staged batched GEMM (64x64 block tile, 8 waves) -------
// A row-major [M x Kd]; B normal [Kd x N] (OPB=0) or transposed [N x Kd]
// (OPB=1). Panels staged per 16-wide K chunk via async load-to-LDS; each
// panel is reused by all 8 waves (4x traffic reduction vs direct).
template<int OPB, int EPI>
__global__ void __launch_bounds__(256) gemm_lds(
    const float* __restrict__ A, long sA, int aMod, int lda,
    const float* __restrict__ Bm, long sB, int bMod, int ldb,
    float* __restrict__ C, long sC, int ldc,
    int M, int Nn, int Kd, int nbatch)
{
  __shared__ float As[64][20];                       // pad: bank-conflict-free
  __shared__ float Bs[OPB ? 64 : 16][OPB ? 20 : 64];
  int tilesM = M >> 6, tilesN = Nn >> 6;
  int per = tilesM * tilesN;
  int bat = blockIdx.x / per;
  int t   = blockIdx.x - bat * per;
  int bm  = t / tilesN;
  int bn  = t - bm * tilesN;
  const float* Ab = A  + (long)(bat % aMod) * sA;
  const float* Bb = Bm + (long)(bat % bMod) * sB;
  float*       Cb = C  + (long)bat * sC;
  int tid = threadIdx.x;
  int w = tid >> 5, lane = tid & 31, half = lane >> 4, lr = lane & 15;
  int tm  = (w >> 1) << 4;        // 0,16,32,48
  int tn2 = (w & 1) << 5;         // 0,32
  int rowBase = bm << 6, colBase = bn << 6;
  int ar = tid >> 2,  ac = (tid & 3) << 2;   // A/Bt staging: 64 rows x 16
  int br = tid >> 4,  bc = (tid & 15) << 2;  // B  staging: 16 rows x 64
  v8f acc0 = {0.f,0.f,0.f,0.f,0.f,0.f,0.f,0.f};
  v8f acc1 = {0.f,0.f,0.f,0.f,0.f,0.f,0.f,0.f};
  for (int k0 = 0; k0 < Kd; k0 += 16) {
    cp16(&As[ar][ac], &Ab[(long)(rowBase + ar) * lda + k0 + ac]);
    if (OPB) cp16(&Bs[ar][ac], &Bb[(long)(colBase + ar) * ldb + k0 + ac]);
    else     cp16(&Bs[br][bc], &Bb[(long)(k0 + br) * ldb + colBase + bc]);
    ASYNC_WAIT();
    __syncthreads();
#pragma unroll
    for (int kl = 0; kl < 16; kl += 4) {
      int kk = kl + (half << 1);
      v2f av = *(const v2f*)&As[tm + lr][kk];
      v2f b0, b1;
      if (OPB) {
        b0 = *(const v2f*)&Bs[tn2 + lr][kk];
        b1 = *(const v2f*)&Bs[tn2 + 16 + lr][kk];
      } else {
        b0.x = Bs[kk][tn2 + lr];      b0.y = Bs[kk + 1][tn2 + lr];
        b1.x = Bs[kk][tn2 + 16 + lr]; b1.y = Bs[kk + 1][tn2 + 16 + lr];
      }
      acc0 = __builtin_amdgcn_wmma_f32_16x16x4_f32(false, av, false, b0,
                                                   (short)0, acc0, false, false);
      acc1 = __builtin_amdgcn_wmma_f32_16x16x4_f32(false, av, false, b1,
                                                   (short)0, acc1, false, false);
    }
    __syncthreads();
  }
  wmma_store<EPI>(Cb, ldc, rowBase + tm, colBase + tn2, acc0);
  wmma_store<EPI>(Cb, ldc, rowBase + tm, colBase + tn2 + 16, acc1);
}

// ------------------------------------------------- landmark adjacency -> an
__global__ void build_an_k(const float* __restrict__ elems, float* __restrict__ an)
{
  int l = blockIdx.x;
  __shared__ float Ab[cK * cK];
  __shared__ float dn[cK];
  const float* el = elems + (long)l * (cK * (cK - 1) / 2);
  for (int idx = threadIdx.x; idx < cK * cK; idx += blockDim.x) {
    int i = idx / cK, j = idx % cK;
    float v = 0.f;
    if (i != j) {
      int a = i < j ? i : j;
      int b = i < j ? j : i;
      int f = a * cK - (a * (a + 1)) / 2 + (b - a - 1);   // triu_indices(K,1)
      v = (el[f] != 0.f) ? 1.f : 0.f;                      // nonzero -> binary
    }
    Ab[idx] = v;
  }
  __syncthreads();
  if (threadIdx.x < cK) {
    float s = 1.f; // self loop
    for (int j = 0; j < cK; ++j) s += Ab[threadIdx.x * cK + j];
    dn[threadIdx.x] = rsqrtf(s);
  }
  __syncthreads();
  for (int idx = threadIdx.x; idx < cK * cK; idx += blockDim.x) {
    int i = idx / cK, j = idx % cK;
    float ah = Ab[idx] + (i == j ? 1.f : 0.f);
    an[(long)l * cK * cK + idx] = dn[i] * ah * dn[j];
  }
}

// ------------------------------------------------------ attention scores
__global__ void scores_k(const float* __restrict__ cw, const float* __restrict__ hw,
                         const float* __restrict__ alpha,
                         float* __restrict__ s_c, float* __restrict__ s_o)
{
  int idx = blockIdx.x * blockDim.x + threadIdx.x;
  if (idx < cL * cK) {
    const float* row = cw + (long)idx * cD;
    float s = 0.f;
    for (int d = 0; d < cD; ++d) {
      float v = row[d];
      v = v > 0.f ? v : 0.2f * v;
      s += v * alpha[d];
    }
    s_c[idx] = s;
  }
  int t = idx - cL * cK;
  if (t >= 0 && t < cT) {
    const float* row = hw + (long)t * cD;
    float s = 0.f;
    for (int d = 0; d < cD; ++d) {
      float v = row[d];
      v = v > 0.f ? v : 0.2f * v;
      s += v * alpha[cD + d];
    }
    s_o[t] = s;
  }
}

// ------------------------- softmax(sigmoid(S)) with torch reshape scramble
__global__ void softmax_E_k(const float* __restrict__ s_c, const float* __restrict__ s_o,
                            float* __restrict__ E)
{
  __shared__ float red[4][64];
  int grp = threadIdx.x >> 6;
  int k   = threadIdx.x & 63;
  long row = (long)blockIdx.x * 4 + grp;   // < cL*cT
  int l  = (int)(row / cT);
  int tp = (int)(row % cT);
  float v = s_c[l * cK + (tp >> 8)] + s_o[((tp & 255) << 6) + k];
  v = 1.f / (1.f + expf(-v));              // sigmoid
  red[grp][k] = v; __syncthreads();
  for (int s = 32; s > 0; s >>= 1) {
    if (k < s) red[grp][k] = fmaxf(red[grp][k], red[grp][k + s]);
    __syncthreads();
  }
  float mx = red[grp][0]; __syncthreads();
  float ev = expf(v - mx);
  red[grp][k] = ev; __syncthreads();
  for (int s = 32; s > 0; s >>= 1) {
    if (k < s) red[grp][k] += red[grp][k + s];
    __syncthreads();
  }
  E[row * cK + k] = ev / red[grp][0];
}

__global__ void colsum_k(const float* __restrict__ E, float* __restrict__ cs)
{
  int lb = blockIdx.x;         // < cL*cB
  int k  = threadIdx.x;        // blockDim = 64
  const float* base = E + (long)lb * cN * cK + k;
  float s = 0.f;
  for (int n = 0; n < cN; ++n) s += base[(long)n * cK];
  cs[lb * cK + k] = s;
}

__global__ void en_k(const float* __restrict__ E, const float* __restrict__ cs,
                     float* __restrict__ En)
{
  long idx = (long)blockIdx.x * blockDim.x + threadIdx.x;  // < cL*cB*cN*cK
  long lb  = idx / ((long)cN * cK);
  int  k   = (int)(idx % cK);
  En[idx] = E[idx] / cs[lb * cK + k];
}

// ---------------------------- normalized symmetric Laplacian (160 matrices)
__global__ void lap_k(float* __restrict__ rev, const float* __restrict__ adj,
                      float* __restrict__ lapA)
{
  int blk = blockIdx.x;
  const float* src;
  float* dst;
  if (blk < cL * cB) { src = rev + (long)blk * cN * cN; dst = rev + (long)blk * cN * cN; }
  else { src = adj + (long)(blk - cL * cB) * cN * cN; dst = lapA + (long)(blk - cL * cB) * cN * cN; }
  __shared__ float dn[cN];
  for (int i = threadIdx.x; i < cN; i += blockDim.x) {
    float s = 0.f;                                  // column sums (adj.sum(-2))
    for (int r = 0; r < cN; ++r) s += src[(long)r * cN + i];
    dn[i] = rsqrtf(s);
  }
  __syncthreads();
  for (int idx = threadIdx.x; idx < cN * cN; idx += blockDim.x) {
    int i = idx / cN, j = idx % cN;
    if (j < i) continue;                            // one thread per (i<=j)
    float mij = src[(long)i * cN + j];
    float mji = src[(long)j * cN + i];
    float dg  = (i == j) ? 1.f : 0.f;
    float v = 0.5f * ((dg - dn[i] * mij * dn[j]) + (dg - dn[j] * mji * dn[i]));
    dst[(long)i * cN + j] = v;
    dst[(long)j * cN + i] = v;
  }
}

// ------------- symmetric eigenvalues: Householder tridiag + tql1 + sort
__global__ void eig_k(float* __restrict__ rev, float* __restrict__ lapA,
                      float* __restrict__ eigR, float* __restrict__ eigA)
{
  int blk = blockIdx.x;
  float* A;
  float* out;
  if (blk < cL * cB) { A = rev + (long)blk * cN * cN; out = eigR + (long)blk * cN; }
  else { A = lapA + (long)(blk - cL * cB) * cN * cN; out = eigA + (long)(blk - cL * cB) * cN; }

  __shared__ float u[cN], p[cN], d[cN], e[cN];
  __shared__ float red[256];
  __shared__ float shH;
  int tid = threadIdx.x;

  for (int i = cN - 1; i >= 2; --i) {
    float ls = 0.f;
    for (int k2 = tid; k2 < i; k2 += 256) {
      float v = A[(long)i * cN + k2];
      u[k2] = v;
      ls += fabsf(v);
    }
    float scale = blk_sum256(ls, red);
    if (scale == 0.f) {
      if (tid == 0) e[i] = 0.f;
      __syncthreads();
      continue;
    }
    float hs = 0.f;
    for (int k2 = tid; k2 < i; k2 += 256) {
      float v = u[k2] / scale;
      u[k2] = v;
      hs += v * v;
    }
    float h = blk_sum256(hs, red);
    if (tid == 0) {
      float f = u[i - 1];
      float g = (f >= 0.f) ? -sqrtf(h) : sqrtf(h);
      e[i] = scale * g;
      shH = h - f * g;
      u[i - 1] = f - g;
    }
    __syncthreads();
    float hh = shH;
    for (int j = tid; j < i; j += 256) {           // p = A*u / h
      const float* row = A + (long)j * cN;
      float s = 0.f;
      for (int k2 = 0; k2 < i; ++k2) s += row[k2] * u[k2];
      p[j] = s / hh;
    }
    __syncthreads();
    float ds = 0.f;
    for (int k2 = tid; k2 < i; k2 += 256) ds += u[k2] * p[k2];
    float Kc = blk_sum256(ds, red) / (2.f * hh);
    for (int k2 = tid; k2 < i; k2 += 256) p[k2] -= Kc * u[k2];  // q in p
    __syncthreads();
    for (int j = tid; j < i; j += 256) {           // A -= u q^T + q u^T
      float uj = u[j], qj = p[j];
      float* row = A + (long)j * cN;
      for (int k2 = 0; k2 < i; ++k2) row[k2] -= uj * p[k2] + qj * u[k2];
    }
    __syncthreads();
  }
  if (tid == 0) { e[1] = A[cN]; e[0] = 0.f; }      // A[1][0]
  for (int j = tid; j < cN; j += 256) d[j] = A[(long)j * cN + j];
  __syncthreads();

  if (tid == 0) {
    // tql1: implicit QL with shifts (eigenvalues only)
    for (int i = 1; i < cN; ++i) e[i - 1] = e[i];
    e[cN - 1] = 0.f;
    for (int l2 = 0; l2 < cN; ++l2) {
      int iter = 0, m;
      do {
        for (m = l2; m < cN - 1; ++m) {
          float dd = fabsf(d[m]) + fabsf(d[m + 1]);
          if (fabsf(e[m]) <= 1.1920929e-07f * dd) break;
        }
        if (m != l2) {
          if (iter++ == 60) break;
          float g = (d[l2 + 1] - d[l2]) / (2.f * e[l2]);
          float r = hypotf(g, 1.f);
          g = d[m] - d[l2] + e[l2] / (g + copysignf(r, g));
          float s = 1.f, c = 1.f, pp = 0.f;
          int i2, early = 0;
          r = 0.f;
          for (i2 = m - 1; i2 >= l2; --i2) {
            float f  = s * e[i2];
            float b2 = c * e[i2];
            r = hypotf(f, g);
            e[i2 + 1] = r;
            if (r == 0.f) { d[i2 + 1] -= pp; e[m] = 0.f; early = 1; break; }
            s = f / r;
            c = g / r;
            g = d[i2 + 1] - pp;
            r = (d[i2] - g) * s + 2.f * c * b2;
            pp = s * r;
            d[i2 + 1] = g + pp;
            g = c * r - b2;
          }
          if (early) continue;
          d[l2] -= pp; e[l2] = g; e[m] = 0.f;
        }
      } while (m != l2);
    }
    for (int i = 1; i < cN; ++i) {                 // sort ascending
      float key = d[i];
      int j = i - 1;
      while (j >= 0 && d[j] > key) { d[j + 1] = d[j]; --j; }
      d[j + 1] = key;
    }
  }
  __syncthreads();
  for (int j = tid; j < cN; j += 256) out[j] = d[j];
}

__global__ void sp_k(const float* __restrict__ eigR, const float* __restrict__ eigA,
                     float* __restrict__ sp)
{
  int lb = blockIdx.x;       // < cL*cB
  int b  = lb % cB;
  __shared__ float red[256];
  float s = 0.f;
  for (int i = threadIdx.x; i < cN; i += 256)
    s += fabsf(eigA[(long)b * cN + i] - eigR[(long)lb * cN + i]);
  float tot = blk_sum256(s, red);
  if (threadIdx.x == 0) sp[lb] = tot;
}

__global__ void jmin_k(const float* __restrict__ sp, int* __restrict__ jmin,
                       float* __restrict__ spmin_out)
{
  int b = threadIdx.x;
  if (b >= cB) return;
  float best = sp[b];
  int bi = 0;
  for (int l = 1; l < cL; ++l) {
    float v = sp[l * cB + b];
    if (v < best) { best = v; bi = l; }    // first min wins (argmin)
  }
  jmin[b] = bi;
  spmin_out[b] = best;
}

__global__ void poolsel_k(const float* __restrict__ pool, const int* __restrict__ jmin,
                          float* __restrict__ out)
{
  int b = blockIdx.x;
  const float* src = pool + ((long)jmin[b] * cB + b) * cK * cK;
  float* dst = out + (long)b * cK * cK;
  for (int i = threadIdx.x; i < cK * cK; i += blockDim.x) dst[i] = src[i];
}

// feat[b] = E_sel^T (KxN) @ h[b] (NxD) via fp32 WMMA
__global__ void feat_k(const float* __restrict__ E, const float* __restrict__ h,
                       const int* __restrict__ jmin, float* __restrict__ out)
{
  int wid = blockIdx.x * (blockDim.x >> 5) + (threadIdx.x >> 5);
  const int tilesN = cD / 16;                  // 8
  const int per = (cK / 16) * tilesN;          // 32
  if (wid >= per * cB) return;
  int b  = wid / per;
  int t  = wid % per;
  int tm = t / tilesN, tn = t % tilesN;
  const float* Ab = E + ((long)jmin[b] * cB + b) * (long)cN * cK;  // [N,K], used transposed
  const float* Bb = h + (long)b * cN * cD;
  v8f acc = wmma_mm<1, 0>(Ab, cK, Bb, cD, tm << 4, tn << 4, cN);
  wmma_store<0>(out + (long)b * cK * cD, cD, tm << 4, tn << 4, acc);
}

// ---------------------------------------------------------------- launcher
static inline int gblk(long m, long n, long nb) {
  long waves = (m / 16) * (n / 16) * nb;
  return (int)((waves + 7) / 8);
}
static inline int gblk64(long m, long n, long nb) {
  return (int)((m / 64) * (n / 64) * nb);
}

extern "C" void kernel_launch(void* const* d_in, const int* in_sizes, int n_in,
                              void* d_out, int out_size, void* d_ws, size_t ws_size,
                              hipStream_t stream)
{
  (void)in_sizes; (void)n_in; (void)out_size; (void)ws_size;
  const float* h     = (const float*)d_in[0];
  const float* adj   = (const float*)d_in[1];
  const float* lmel  = (const float*)d_in[2];
  const float* W0    = (const float*)d_in[3];
  const float* W1    = (const float*)d_in[4];
  const float* Wa    = (const float*)d_in[5];
  const float* alpha = (const float*)d_in[6];
  float* out = (float*)d_out;
  float* ws  = (float*)d_ws;

  float* an   = ws;
  float* x1   = an   + (long)cL * cK * cK;
  float* x2   = x1   + (long)cL * cK * cD;
  float* coar = x2   + (long)cL * cK * cD;
  float* cw   = coar + (long)cL * cK * cD;
  float* hw   = cw   + (long)cL * cK * cD;
  float* s_c  = hw   + (long)cT * cD;
  float* s_o  = s_c  + (long)cL * cK;
  float* E    = s_o  + cT;
  float* cs   = E    + (long)cL * cT * cK;
  float* AS   = cs   + (long)cL * cB * cK;
  float* pool = AS   + (long)cL * cT * cK;
  float* En   = pool + (long)cL * cB * cK * cK;
  float* EnP  = AS;                               // reuse (AS dead after pool)
  float* rev  = En   + (long)cL * cT * cK;
  float* lapA = rev  + (long)cL * cB * cN * cN;
  float* eigR = lapA + (long)cB * cN * cN;
  float* eigA = eigR + (long)cL * cB * cN;
  float* sp   = eigA + (long)cB * cN;
  int*   jmin = (int*)(sp + cL * cB);

  // 1) landmark normalized adjacency an[L,K,K]
  build_an_k<<<cL, 256, 0, stream>>>(lmel, an);
  // 2) GCN: x1 = relu(an@W0), x2 = x1@W1, coar = an@x2
  gemm_f32<0,0,1><<<gblk(cK, cD, cL), 256, 0, stream>>>(
      an, (long)cK * cK, cL, cK, W0, 0L, 1, cD, x1, (long)cK * cD, cD, cK, cD, cK, cL);
  gemm_f32<0,0,0><<<gblk(cL * cK, cD, 1), 256, 0, stream>>>(
      x1, 0L, 1, cD, W1, 0L, 1, cD, x2, 0L, cD, cL * cK, cD, cD, 1);
  gemm_f32<0,0,0><<<gblk(cK, cD, cL), 256, 0, stream>>>(
      an, (long)cK * cK, cL, cK, x2, (long)cK * cD, cL, cD, coar, (long)cK * cD, cD, cK, cD, cK, cL);
  // 3) attention projections
  gemm_f32<0,0,0><<<gblk(cL * cK, cD, 1), 256, 0, stream>>>(
      coar, 0L, 1, cD, Wa, 0L, 1, cD, cw, 0L, cD, cL * cK, cD, cD, 1);
  gemm_lds<0,0><<<gblk64(cT, cD, 1), 256, 0, stream>>>(
      h, 0L, 1, cD, Wa, 0L, 1, cD, hw, 0L, cD, cT, cD, cD, 1);
  scores_k<<<(cL * cK + cT + 255) / 256, 256, 0, stream>>>(cw, hw, alpha, s_c, s_o);
  // 4) assignments E, column sums
  softmax_E_k<<<(cL * cT) / 4, 256, 0, stream>>>(s_c, s_o, E);
  colsum_k<<<cL * cB, 64, 0, stream>>>(E, cs);
  // 5) AS = adj @ E   [L*B batched 512x512x64] — LDS-staged
  gemm_lds<0,0><<<gblk64(cN, cK, cL * cB), 256, 0, stream>>>(
      adj, (long)cN * cN, cB, cN, E, (long)cN * cK, cL * cB, cK,
      AS, (long)cN * cK, cK, cN, cK, cN, cL * cB);
  // 6) pool = E^T @ AS, zero diag (A transposed -> direct kernel)
  gemm_f32<1,0,2><<<gblk(cK, cK, cL * cB), 256, 0, stream>>>(
      E, (long)cN * cK, cL * cB, cK, AS, (long)cN * cK, cL * cB, cK,
      pool, (long)cK * cK, cK, cK, cK, cN, cL * cB);
  // 7) En, EnP = En@pool, rev = EnP@En^T (zero diag) — LDS-staged
  en_k<<<(int)(((long)cL * cB * cN * cK) / 256), 256, 0, stream>>>(E, cs, En);
  gemm_lds<0,0><<<gblk64(cN, cK, cL * cB), 256, 0, stream>>>(
      En, (long)cN * cK, cL * cB, cK, pool, (long)cK * cK, cL * cB, cK,
      EnP, (long)cN * cK, cK, cN, cK, cK, cL * cB);
  gemm_lds<1,2><<<gblk64(cN, cN, cL * cB), 256, 0, stream>>>(
      EnP, (long)cN * cK, cL * cB, cK, En, (long)cN * cK, cL * cB, cK,
      rev, (long)cN * cN, cN, cN, cN, cK, cL * cB);
  // 8) Laplacians, eigenvalues, spectral distances
  lap_k<<<cL * cB + cB, 256, 0, stream>>>(rev, adj, lapA);
  eig_k<<<cL * cB + cB, 256, 0, stream>>>(rev, lapA, eigR, eigA);
  sp_k<<<cL * cB, 256, 0, stream>>>(eigR, eigA, sp);
  // 9) selection + outputs
  jmin_k<<<1, 32, 0, stream>>>(sp, jmin, out + SPMIN_OFF);
  poolsel_k<<<cB, 256, 0, stream>>>(pool, jmin, out + POOLSEL_OFF);
  feat_k<<<gblk(cK, cD, cB), 256, 0, stream>>>(E, h, jmin, out);
}